// Attention_75874892251845
// MI455X (gfx1250) — compile-verified
//
#include <hip/hip_runtime.h>
#include <cstddef>
#include <cstdint>

typedef __attribute__((ext_vector_type(2))) float v2f;
typedef __attribute__((ext_vector_type(8))) float v8f;

#define WMMA_F32(a, b, c) \
  __builtin_amdgcn_wmma_f32_16x16x4_f32(false, (a), false, (b), (short)0, (c), false, false)

// ---- CDNA5 async global->LDS copy (ASYNCcnt) with safe fallback ------------
#if defined(__has_builtin)
#if __has_builtin(__builtin_amdgcn_global_load_async_to_lds_b128)
#define HAVE_ASYNC_LDS 1
#endif
#endif
#ifndef HAVE_ASYNC_LDS
#define HAVE_ASYNC_LDS 0
#endif

#if HAVE_ASYNC_LDS
// Builtin signature (from hipcc diagnostic): param0 is
// 'int __attribute__((vector_size(16))) __device__ *' i.e. int4 in AS1;
// param1 is the LDS-side int4 pointer (AS3); then imm offset, imm cpol.
typedef int v4i_gcc __attribute__((vector_size(16)));
typedef v4i_gcc __attribute__((address_space(1)))* gv4_ptr;
typedef v4i_gcc __attribute__((address_space(3)))* lv4_ptr;

__device__ __forceinline__ void async_copy_b128(const float* g, float* l) {
  gv4_ptr gp = reinterpret_cast<gv4_ptr>(reinterpret_cast<uintptr_t>(g));
  // AS3 pointers are 32-bit; generic LDS addresses carry the offset in the low bits.
  lv4_ptr lp = reinterpret_cast<lv4_ptr>((unsigned int)reinterpret_cast<uintptr_t>(l));
  __builtin_amdgcn_global_load_async_to_lds_b128(gp, lp, 0, 0);
}
#endif

__device__ __forceinline__ void wait_async() {
#if HAVE_ASYNC_LDS
#if __has_builtin(__builtin_amdgcn_s_wait_asynccnt)
  __builtin_amdgcn_s_wait_asynccnt(0);
#else
  asm volatile("s_wait_asynccnt 0" ::: "memory");
#endif
#endif
}

// Stage one 16-byte chunk: async DMA if available, else load+store.
__device__ __forceinline__ void stage16(const float* g, float* l) {
#if HAVE_ASYNC_LDS
  async_copy_b128(g, l);
#else
  float4 t = *reinterpret_cast<const float4*>(g);
  l[0] = t.x; l[1] = t.y; l[2] = t.z; l[3] = t.w;
#endif
}

// ---------------------------------------------------------------------------
// GEMM: C[M,N] = A[M,K] * B[K,N], all row-major fp32.
// Requires M%64==0, N%128==0, K%16==0 (true for all shapes here).
// Block: 256 threads (8 waves). Block tile 64x128, BK=16, double-buffered LDS
// with async global->LDS staging overlapping the WMMA work.
// Wave tile 32x32 = 2x2 WMMA_F32_16x16x4 accumulators.
// ---------------------------------------------------------------------------
__global__ __launch_bounds__(256) void gemm_f32_wmma(const float* __restrict__ A,
                                                     const float* __restrict__ B,
                                                     float* __restrict__ C,
                                                     int M, int N, int K) {
  constexpr int SA = 36;   // row stride: 144 B = 9*16 -> B128-aligned LDS writes
  constexpr int SB = 132;  // row stride: 528 B = 33*16
  __shared__ float As[2][64 * SA];
  __shared__ float Bs[2][16 * SB];

  const int tid  = threadIdx.x;
  const int lane = tid & 31;
  const int wave = tid >> 5;
  const int half = lane >> 4;   // 0/1: which 16-lane half
  const int l16  = lane & 15;
  const int wm   = (wave >> 2) * 32;  // wave M offset: 0 or 32
  const int wn   = (wave & 3) * 32;   // wave N offset: 0..96

  const int n0 = blockIdx.x * 128;
  const int m0 = blockIdx.y * 64;

  const int ar = tid >> 2;        // 0..63 : A tile row
  const int ac = (tid & 3) * 4;   // 0..12 : A tile col (float4)
  const int br0 = tid >> 5;       // 0..7  : B tile rows for i=0
  const int bc  = (tid & 31) * 4; // 0..124

  v8f acc[2][2];
#pragma unroll
  for (int mi = 0; mi < 2; ++mi)
#pragma unroll
    for (int ni = 0; ni < 2; ++ni)
#pragma unroll
      for (int v = 0; v < 8; ++v) acc[mi][ni][v] = 0.0f;

  // Stage tile (A: 64x16, B: 16x128) for k-offset kt into buffer buf.
  auto stage_tile = [&](int buf, int kt) {
    stage16(A + (size_t)(m0 + ar) * K + kt + ac, &As[buf][ar * SA + ac]);
#pragma unroll
    for (int i = 0; i < 2; ++i) {
      int brr = br0 + i * 8;  // 0..15
      stage16(B + (size_t)(kt + brr) * N + n0 + bc, &Bs[buf][brr * SB + bc]);
    }
  };

  stage_tile(0, 0);
  wait_async();
  __syncthreads();

  int cur = 0;
  for (int kt = 0; kt < K; kt += 16) {
    const int nxt = kt + 16;
    if (nxt < K) stage_tile(cur ^ 1, nxt);  // DMA next tile while computing

    const float* Ac = &As[cur][0];
    const float* Bc = &Bs[cur][0];
#pragma unroll
    for (int kk = 0; kk < 16; kk += 4) {
      v2f af[2], bf[2];
#pragma unroll
      for (int mi = 0; mi < 2; ++mi) {
        // A frag: lane row = l16, K = kk + 2*half + {0,1}
        int r = wm + mi * 16 + l16;
        af[mi].x = Ac[r * SA + kk + 2 * half];
        af[mi].y = Ac[r * SA + kk + 2 * half + 1];
      }
#pragma unroll
      for (int ni = 0; ni < 2; ++ni) {
        // B frag: lane col = l16, K = kk + 2*half + {0,1}
        int c  = wn + ni * 16 + l16;
        int kr = kk + 2 * half;
        bf[ni].x = Bc[kr * SB + c];
        bf[ni].y = Bc[(kr + 1) * SB + c];
      }
#pragma unroll
      for (int mi = 0; mi < 2; ++mi)
#pragma unroll
        for (int ni = 0; ni < 2; ++ni)
          acc[mi][ni] = WMMA_F32(af[mi], bf[ni], acc[mi][ni]);
    }

    if (nxt < K) wait_async();  // next buffer landed (this wave's DMAs)
    __syncthreads();            // everyone done reading cur + DMAs visible
    cur ^= 1;
  }

  // Epilogue: C layout -> row = v + 8*half, col = l16 within each 16x16 tile.
#pragma unroll
  for (int mi = 0; mi < 2; ++mi)
#pragma unroll
    for (int ni = 0; ni < 2; ++ni)
#pragma unroll
      for (int v = 0; v < 8; ++v) {
        int row = m0 + wm + mi * 16 + v + 8 * half;
        int col = n0 + wn + ni * 16 + l16;
        C[(size_t)row * N + col] = acc[mi][ni][v];
      }
}

// ---------------------------------------------------------------------------
// RoPE (interleaved pairs), in-place on [B*S, H, 128] buffer.
// ---------------------------------------------------------------------------
__global__ __launch_bounds__(256) void rope_f32(float* __restrict__ x,
                                                const float* __restrict__ cs,
                                                const float* __restrict__ sn,
                                                int H) {
  int idx  = blockIdx.x * 256 + threadIdx.x;
  int p    = idx & 63;          // pair index 0..63
  int rest = idx >> 6;          // (b*S + s)*H + h
  int bs   = rest / H;
  int s    = bs & 2047;         // S = 2048
  float c  = cs[s * 64 + p];
  float sv = sn[s * 64 + p];
  float* ptr = x + (size_t)rest * 128 + 2 * p;
  float xr = ptr[0];
  float xi = ptr[1];
  ptr[0] = xr * c - xi * sv;
  ptr[1] = xr * sv + xi * c;
}

// ---------------------------------------------------------------------------
// Flash attention (causal; WINDOW==SEQLEN so only j<=i matters).
// Grid: B*H*(S/128) blocks of 256 threads. Wave w handles 16 query rows.
// K/V staged in LDS (async DMA) in chunks of 32 keys, shared by all 8 waves.
// ---------------------------------------------------------------------------
__global__ __launch_bounds__(256) void attn_f32_wmma(const float* __restrict__ Q,
                                                     const float* __restrict__ Kb,
                                                     const float* __restrict__ Vb,
                                                     float* __restrict__ O) {
  constexpr int S = 2048, H = 32, KVH = 8, HD = 128;
  constexpr int KC = 32;                         // keys per chunk
  constexpr int KSTR = 132, VSTR = 132, PSTR = 34;
  constexpr float SCALE = 0.08838834764831845f;  // 128^-0.5
  __shared__ float Ks[KC * KSTR];                // 16.5 KB (528 B rows, B128-aligned)
  __shared__ float Vs[KC * VSTR];                // 16.5 KB
  __shared__ float Ps[8 * 16 * PSTR];            // 17.0 KB (per-wave P scratch)

  const int tid  = threadIdx.x;
  const int lane = tid & 31;
  const int wave = tid >> 5;
  const int half = lane >> 4;
  const int l16  = lane & 15;

  const int qt  = blockIdx.x & 15;   // S/128 = 16 query tiles
  const int bh  = blockIdx.x >> 4;
  const int h   = bh & (H - 1);
  const int b   = bh >> 5;
  const int kvh = h >> 2;            // GQA: 4 q-heads per kv-head

  const int qb = qt * 128;
  const int q0 = qb + wave * 16;     // this wave's first query row

  // Q fragments (A layout): 16 rows x 128 d, 32 fragments of v2f.
  v2f qa[32];
  const float* qrow = Q + ((size_t)(b * S + q0 + l16)) * (H * HD) + h * HD;
#pragma unroll
  for (int d4 = 0; d4 < 32; ++d4)
    qa[d4] = *reinterpret_cast<const v2f*>(qrow + 4 * d4 + 2 * half);

  v8f oacc[8];
#pragma unroll
  for (int n = 0; n < 8; ++n)
#pragma unroll
    for (int v = 0; v < 8; ++v) oacc[n][v] = 0.0f;
  float rmax[8], rsum[8];
#pragma unroll
  for (int v = 0; v < 8; ++v) { rmax[v] = -1e30f; rsum[v] = 0.0f; }

  float* Pw = Ps + wave * 16 * PSTR;
  const int nchunks = (qb + 128) / KC;  // causal upper bound for this block

  for (int ch = 0; ch < nchunks; ++ch) {
    const int kb = ch * KC;
    // Stage K/V chunk (32 rows x 128 each): 4 B128 per thread per tensor.
#pragma unroll
    for (int i = 0; i < 4; ++i) {
      int id = tid + i * 256;   // 0..1023
      int r  = id >> 5;         // 0..31
      int c  = (id & 31) * 4;   // 0..124
      size_t goff = (size_t)(b * S + kb + r) * (KVH * HD) + kvh * HD + c;
      stage16(Kb + goff, &Ks[r * KSTR + c]);
      stage16(Vb + goff, &Vs[r * VSTR + c]);
    }
    wait_async();
    __syncthreads();

    const bool active = (kb <= q0 + 15);  // chunk has any unmasked key for us
    if (active) {
      // S = Q * K^T : two 16x16 tiles, 32 k-steps over HD.
      v8f sacc[2];
#pragma unroll
      for (int t = 0; t < 2; ++t)
#pragma unroll
        for (int v = 0; v < 8; ++v) sacc[t][v] = 0.0f;
#pragma unroll
      for (int d4 = 0; d4 < 32; ++d4) {
#pragma unroll
        for (int t = 0; t < 2; ++t) {
          // B = K^T: B[d][key] = Ks[key][d]; lane key = l16, d = 4*d4+2*half+{0,1}
          const float* kp = Ks + (t * 16 + l16) * KSTR + 4 * d4 + 2 * half;
          v2f bf;
          bf.x = kp[0];
          bf.y = kp[1];
          sacc[t] = WMMA_F32(qa[d4], bf, sacc[t]);
        }
      }

      // Scale + causal mask + online softmax (rows live in 16-lane halves).
      const bool need_mask = (kb + KC - 1) > q0;
#pragma unroll
      for (int v = 0; v < 8; ++v) {
        const int irow = q0 + v + 8 * half;
        float s0 = sacc[0][v] * SCALE;
        float s1 = sacc[1][v] * SCALE;
        if (need_mask) {
          if (kb + l16 > irow)      s0 += -1.0e9f;
          if (kb + 16 + l16 > irow) s1 += -1.0e9f;
        }
        float tmax = fmaxf(s0, s1);
#pragma unroll
        for (int off = 1; off < 16; off <<= 1)
          tmax = fmaxf(tmax, __shfl_xor(tmax, off, 32));
        float mnew = fmaxf(rmax[v], tmax);
        float corr = __expf(rmax[v] - mnew);
        rmax[v] = mnew;
        float p0 = __expf(s0 - mnew);
        float p1 = __expf(s1 - mnew);
        float psum = p0 + p1;
#pragma unroll
        for (int off = 1; off < 16; off <<= 1)
          psum += __shfl_xor(psum, off, 32);
        rsum[v] = rsum[v] * corr + psum;
#pragma unroll
        for (int n = 0; n < 8; ++n) oacc[n][v] *= corr;
        const int r = v + 8 * half;        // C-layout row -> LDS (A-layout source)
        Pw[r * PSTR + l16]      = p0;
        Pw[r * PSTR + 16 + l16] = p1;
      }

      // O += P * V : A = P (16x32) from per-wave LDS, B = Vs rows.
#pragma unroll
      for (int kk = 0; kk < KC; kk += 4) {
        v2f pf;
        pf.x = Pw[l16 * PSTR + kk + 2 * half];
        pf.y = Pw[l16 * PSTR + kk + 2 * half + 1];
        int kr = kk + 2 * half;
#pragma unroll
        for (int n = 0; n < 8; ++n) {
          v2f bv;
          bv.x = Vs[kr * VSTR + n * 16 + l16];
          bv.y = Vs[(kr + 1) * VSTR + n * 16 + l16];
          oacc[n] = WMMA_F32(pf, bv, oacc[n]);
        }
      }
    }
    __syncthreads();
  }

  // Write attention output in [b*S, H*HD] layout for the final projection.
#pragma unroll
  for (int n = 0; n < 8; ++n)
#pragma unroll
    for (int v = 0; v < 8; ++v) {
      int row = q0 + v + 8 * half;
      O[((size_t)(b * S + row)) * (H * HD) + h * HD + n * 16 + l16] =
          oacc[n][v] / rsum[v];
    }
}

// ---------------------------------------------------------------------------
// Launcher. Inputs (setup_inputs order):
// 0:x 1:freqs_cos 2:freqs_sin 3:positions(unused) 4:mask(unused, recomputed)
// 5:wq 6:wk 7:wv 8:wo
// ---------------------------------------------------------------------------
extern "C" void kernel_launch(void* const* d_in, const int* in_sizes, int n_in,
                              void* d_out, int out_size, void* d_ws, size_t ws_size,
                              hipStream_t stream) {
  (void)in_sizes; (void)n_in; (void)out_size; (void)ws_size;
  const float* x  = (const float*)d_in[0];
  const float* fc = (const float*)d_in[1];
  const float* fs = (const float*)d_in[2];
  const float* wq = (const float*)d_in[5];
  const float* wk = (const float*)d_in[6];
  const float* wv = (const float*)d_in[7];
  const float* wo = (const float*)d_in[8];
  float* out = (float*)d_out;

  const int BS  = 2 * 2048;   // batch*seq rows
  const int DIM = 4096;

  float* q  = (float*)d_ws;                      // [4096, 4096]
  float* k  = q + (size_t)BS * DIM;              // [4096, 1024]
  float* v  = k + (size_t)BS * 1024;             // [4096, 1024]
  float* ao = v + (size_t)BS * 1024;             // [4096, 4096]

  dim3 blk(256);
  gemm_f32_wmma<<<dim3(DIM / 128, BS / 64), blk, 0, stream>>>(x, wq, q, BS, DIM, DIM);
  gemm_f32_wmma<<<dim3(1024 / 128, BS / 64), blk, 0, stream>>>(x, wk, k, BS, 1024, DIM);
  gemm_f32_wmma<<<dim3(1024 / 128, BS / 64), blk, 0, stream>>>(x, wv, v, BS, 1024, DIM);

  rope_f32<<<dim3((BS * 32 * 64) / 256), blk, 0, stream>>>(q, fc, fs, 32);
  rope_f32<<<dim3((BS * 8 * 64) / 256), blk, 0, stream>>>(k, fc, fs, 8);

  attn_f32_wmma<<<dim3(2 * 32 * 16), blk, 0, stream>>>(q, k, v, ao);

  gemm_f32_wmma<<<dim3(DIM / 128, BS / 64), blk, 0, stream>>>(ao, wo, out, BS, DIM, DIM);
}